// CaptionModel_8821862826608
// MI455X (gfx1250) — compile-verified
//
#include <hip/hip_runtime.h>

// Problem constants (from reference): B,K,V,T,S,H
static constexpr int cB = 128;
static constexpr int cK = 10;
static constexpr int cV = 50257;   // = 4*12564 + 1  (tail element IS vocab-1)
static constexpr int cT = 20;
static constexpr int cS = 4;
static constexpr int cH = 2048;

typedef float  f4   __attribute__((ext_vector_type(4), aligned(4)));   // rows only 4B-aligned
typedef float  f4a  __attribute__((ext_vector_type(4), aligned(16)));
typedef unsigned int v4u __attribute__((ext_vector_type(4)));
typedef int    v4i  __attribute__((ext_vector_type(4)));
typedef int    v8i  __attribute__((ext_vector_type(8)));

// ---- order-preserving float key: larger key == larger float; tie -> smaller index wins
__device__ __forceinline__ unsigned flip_f32(float f) {
  unsigned u = __float_as_uint(f);
  return (u & 0x80000000u) ? ~u : (u | 0x80000000u);
}
__device__ __forceinline__ float unflip_f32(unsigned k) {
  unsigned u = (k & 0x80000000u) ? (k ^ 0x80000000u) : ~k;
  return __uint_as_float(u);
}
__device__ __forceinline__ unsigned long long make_key(float v, unsigned idx) {
  return ((unsigned long long)flip_f32(v) << 32) | (unsigned)(~idx);
}

__device__ __forceinline__ void insert_topk(float (&tv)[cK], int (&ti)[cK], float x, int idx) {
  if (x > tv[cK - 1]) {
    tv[cK - 1] = x; ti[cK - 1] = idx;
#pragma unroll
    for (int j = cK - 1; j > 0; --j) {
      if (tv[j] > tv[j - 1]) {
        float tf = tv[j]; tv[j] = tv[j - 1]; tv[j - 1] = tf;
        int   tn = ti[j]; ti[j] = ti[j - 1]; ti[j - 1] = tn;
      }
    }
  }
}

// ============================================================================
// Kernel A: per (b,q) row top-10 over V.  grid = B*K, block = 256 (8 waves).
// Streaming f32x4 loads; last vocab element gets the -1000 penalty.
// ============================================================================
__global__ __launch_bounds__(256) void topk_rows(const float* __restrict__ logprobs,
                                                 float* __restrict__ ws_ysv,
                                                 int*   __restrict__ ws_ixv) {
  const int row = blockIdx.x;            // b*K + q
  const int tid = threadIdx.x;
  const float* rp = logprobs + (size_t)row * cV;

  float tv[cK]; int ti[cK];
#pragma unroll
  for (int j = 0; j < cK; ++j) { tv[j] = -3.402823466e38f; ti[j] = 0x7fffffff; }

  const int n4 = cV >> 2;                // 12564 vec4 groups cover indices [0, V-2]
  const f4* rp4 = (const f4*)rp;
  for (int i = tid; i < n4; i += 256) {
    f4 v = rp4[i];
#pragma unroll
    for (int c = 0; c < 4; ++c)
      insert_topk(tv, ti, v[c], 4 * i + c);
  }
  if (tid == 0) {                        // tail element = vocab-1, penalized
    insert_topk(tv, ti, rp[cV - 1] - 1000.0f, cV - 1);
  }

  // Block merge: 10 rounds of tree-max over per-thread heads (stable on index).
  __shared__ unsigned long long skey[256];
  for (int r = 0; r < cK; ++r) {
    unsigned long long mykey = make_key(tv[0], (unsigned)ti[0]);
    skey[tid] = mykey;
    __syncthreads();
#pragma unroll
    for (int s = 128; s > 0; s >>= 1) {
      if (tid < s) { unsigned long long o = skey[tid + s]; if (o > skey[tid]) skey[tid] = o; }
      __syncthreads();
    }
    unsigned long long best = skey[0];
    if (tid == 0) {
      ws_ysv[row * cK + r] = unflip_f32((unsigned)(best >> 32));
      ws_ixv[row * cK + r] = (int)(~(unsigned)(best & 0xffffffffu));
    }
    if (mykey == best) {                 // winner pops its head (fully unrolled shift)
#pragma unroll
      for (int j = 0; j < cK - 1; ++j) { tv[j] = tv[j + 1]; ti[j] = ti[j + 1]; }
      tv[cK - 1] = -3.402823466e38f; ti[cK - 1] = 0x7fffffff;
    }
    __syncthreads();
  }
}

// ============================================================================
// Kernel B: per-batch candidate ranking + seq/seq_lp rewrite.  grid = B, block = 128.
// flat[f] with f = c*K + q, value = lp_sum[q] + ys[q][c]; stable top-10.
// ============================================================================
__global__ __launch_bounds__(128) void select_and_seq(const float* __restrict__ lpsum,
                                                      const float* __restrict__ ws_ysv,
                                                      const int*   __restrict__ ws_ixv,
                                                      const long long* __restrict__ seq,
                                                      const float* __restrict__ seqlp,
                                                      const int*   __restrict__ tptr,
                                                      long long* __restrict__ out_seq,
                                                      float* __restrict__ out_slp,
                                                      float* __restrict__ out_p,
                                                      int*   __restrict__ ws_qsel) {
  const int b   = blockIdx.x;
  const int tid = threadIdx.x;
  const int t   = *tptr;

  __shared__ unsigned long long skey[128];
  __shared__ int   selq[cK];
  __shared__ int   selc[cK];
  __shared__ float selp[cK];

  float myval = -3.402823466e38f;
  const int myf = tid;                   // flat index
  if (tid < cK * cK) {
    int c = tid / cK, q = tid % cK;
    myval = lpsum[b * cK + q] + ws_ysv[(b * cK + q) * cK + c];
  }

  for (int r = 0; r < cK; ++r) {
    unsigned long long mykey = make_key(myval, (unsigned)myf);
    skey[tid] = mykey;
    __syncthreads();
#pragma unroll
    for (int s = 64; s > 0; s >>= 1) {
      if (tid < s) { unsigned long long o = skey[tid + s]; if (o > skey[tid]) skey[tid] = o; }
      __syncthreads();
    }
    if (mykey == skey[0] && tid < cK * cK) {
      selq[r] = myf % cK; selc[r] = myf / cK; selp[r] = myval;
      myval = -3.402823466e38f;          // consume
    }
    __syncthreads();
  }

  if (tid < cK) {
    const int r = tid, q = selq[r], c = selc[r];
    float rv = ws_ysv[(b * cK + q) * cK + c];
    long long tok = (long long)ws_ixv[(b * cK + q) * cK + c];
    out_p[b * cK + r]   = selp[r];
    ws_qsel[b * cK + r] = q;
    out_seq[((size_t)b * cT + t) * cK + r] = tok;
    out_slp[((size_t)b * cT + t) * cK + r] = rv;
  }
  __syncthreads();

  for (int e = tid; e < cT * cK; e += 128) {
    const int time = e / cK, r = e % cK;
    if (time == t) continue;
    const int src = (time < t) ? selq[r] : r;
    out_seq[((size_t)b * cT + time) * cK + r] = seq[((size_t)b * cT + time) * cK + src];
    out_slp[((size_t)b * cT + time) * cK + r] = seqlp[((size_t)b * cT + time) * cK + src];
  }
}

// ============================================================================
// Kernel C: state permute  new_state[b,s,k,:] = state[b,s,q_sel[b,k],:]
// 8 KB per row; grid = B*S*K, one wave per block.
// CDNA5 path: Tensor Data Mover — tensor_load_to_lds -> s_wait_tensorcnt ->
// tensor_store_from_lds (pure DMA, no VGPR data movement).
// ============================================================================
#if defined(__HIP_DEVICE_COMPILE__) && __has_builtin(__builtin_amdgcn_tensor_load_to_lds) && \
    __has_builtin(__builtin_amdgcn_tensor_store_from_lds)
#define HAVE_TDM 1
#else
#define HAVE_TDM 0
#endif

// --- compile-probe markers (visible in stderr; confirm which path was built) ---
#if HAVE_TDM
#warning "CDNA5-PROBE: TDM builtins available on device pass - TDM path compiled"
#if __clang_major__ >= 23
#warning "CDNA5-PROBE: using 6-arg tensor_load_to_lds (clang >= 23)"
#else
#warning "CDNA5-PROBE: using 5-arg tensor_load_to_lds (clang < 23)"
#endif
#elif defined(__HIP_DEVICE_COMPILE__)
#warning "CDNA5-PROBE: TDM builtins NOT available on device pass - fallback copy in use"
#endif

#if HAVE_TDM
// Build D# group0/group1 for a single contiguous row of cH f32 elements.
__device__ __forceinline__ void tdm_desc_row(unsigned long long gaddr, unsigned lds_byte_off,
                                             v4u& g0, v8i& g1) {
  g0[0] = 1u;                                                  // count=1, user descriptor
  g0[1] = lds_byte_off;                                        // lds_addr
  g0[2] = (unsigned)(gaddr & 0xffffffffu);                     // global_addr[31:0]
  g0[3] = (unsigned)((gaddr >> 32) & 0x1ffffffu) | (2u << 30); // global_addr[56:32] | type=2
  g1[0] = (int)(2u << 16);                                     // data_size=4B; mask=0
  g1[1] = (int)(((unsigned)cH & 0xffffu) << 16);               // tensor_dim0[15:0]
  g1[2] = (int)(((unsigned)cH >> 16) | (1u << 16));            // tensor_dim0[31:16] | tensor_dim1=1
  g1[3] = (int)(((unsigned)cH) << 16);                         // tile_dim0 = H
  g1[4] = 1;                                                   // tile_dim1=1, tile_dim2=0
  g1[5] = (int)cH;                                             // tensor_dim0_stride = H
  g1[6] = 0;
  g1[7] = 0;
}
#endif

__global__ __launch_bounds__(32) void state_gather(const float* __restrict__ state,
                                                   const int*   __restrict__ ws_qsel,
                                                   float*       __restrict__ out_state) {
  const int blk = blockIdx.x;            // (b*S + s)*K + k
  const int k   = blk % cK;
  const int bs  = blk / cK;              // b*S + s
  const int b   = bs / cS;

#if HAVE_TDM
  __shared__ float buf[cH];              // sole LDS object -> group-segment offset 0
  if ((int)blockIdx.x < 0) ((volatile float*)buf)[0] = 0.0f;   // keep LDS allocated

  const int q = __builtin_amdgcn_readfirstlane(ws_qsel[b * cK + k]);
  const float* src = state     + ((size_t)bs * cK + q) * cH;
  float*       dst = out_state + ((size_t)bs * cK + k) * cH;

  v4u g0l; v8i g1l; v4u g0s; v8i g1s;
  v4i z4 = {0, 0, 0, 0};
  tdm_desc_row((unsigned long long)(uintptr_t)src, 0u, g0l, g1l);
  tdm_desc_row((unsigned long long)(uintptr_t)dst, 0u, g0s, g1s);

#if __clang_major__ >= 23
  v8i z8 = {0, 0, 0, 0, 0, 0, 0, 0};
  __builtin_amdgcn_tensor_load_to_lds(g0l, g1l, z4, z4, z8, 0);
#if __has_builtin(__builtin_amdgcn_s_wait_tensorcnt)
  __builtin_amdgcn_s_wait_tensorcnt(0);
#else
  asm volatile("s_wait_tensorcnt 0" ::: "memory");
#endif
  __builtin_amdgcn_tensor_store_from_lds(g0s, g1s, z4, z4, z8, 0);
#else
  __builtin_amdgcn_tensor_load_to_lds(g0l, g1l, z4, z4, 0);
#if __has_builtin(__builtin_amdgcn_s_wait_tensorcnt)
  __builtin_amdgcn_s_wait_tensorcnt(0);
#else
  asm volatile("s_wait_tensorcnt 0" ::: "memory");
#endif
  __builtin_amdgcn_tensor_store_from_lds(g0s, g1s, z4, z4, 0);
#endif
#if __has_builtin(__builtin_amdgcn_s_wait_tensorcnt)
  __builtin_amdgcn_s_wait_tensorcnt(0);
#else
  asm volatile("s_wait_tensorcnt 0" ::: "memory");
#endif

#else  // fallback: vectorized register copy (16B aligned: rows are 8 KB apart)
  const int q = ws_qsel[b * cK + k];
  const f4a* src = (const f4a*)(state     + ((size_t)bs * cK + q) * cH);
  f4a*       dst = (f4a*)      (out_state + ((size_t)bs * cK + k) * cH);
  for (int i = threadIdx.x; i < cH / 4; i += 32) dst[i] = src[i];
#endif
}

// ============================================================================
// Host launcher
// ============================================================================
extern "C" void kernel_launch(void* const* d_in, const int* in_sizes, int n_in,
                              void* d_out, int out_size, void* d_ws, size_t ws_size,
                              hipStream_t stream) {
  (void)in_sizes; (void)n_in; (void)out_size; (void)ws_size;

  const float*     logprobs = (const float*)d_in[0];      // (B,K,V) f32
  const float*     lpsum    = (const float*)d_in[1];      // (B,K)   f32
  const float*     seqlp    = (const float*)d_in[2];      // (B,T,K) f32
  const float*     state    = (const float*)d_in[3];      // (B,S,K,H) f32
  const long long* seq      = (const long long*)d_in[4];  // (B,T,K) i64
  const int*       tptr     = (const int*)d_in[5];        // scalar t

  // Workspace layout
  float* ws_ysv  = (float*)d_ws;                          // B*K*K
  int*   ws_ixv  = (int*)(ws_ysv + (size_t)cB * cK * cK); // B*K*K
  int*   ws_qsel = ws_ixv + (size_t)cB * cK * cK;         // B*K

  // Output layout: outputs concatenated in return order (native dtypes, byte-packed)
  long long* out_seq   = (long long*)d_out;                       // (B,T,K) i64
  float*     out_slp   = (float*)(out_seq + (size_t)cB * cT * cK);// (B,T,K) f32
  float*     out_p     = out_slp + (size_t)cB * cT * cK;          // (B,K)   f32
  float*     out_state = out_p   + (size_t)cB * cK;               // (B,S,K,H) f32

  topk_rows<<<cB * cK, 256, 0, stream>>>(logprobs, ws_ysv, ws_ixv);
  select_and_seq<<<cB, 128, 0, stream>>>(lpsum, ws_ysv, ws_ixv, seq, seqlp, tptr,
                                         out_seq, out_slp, out_p, ws_qsel);
  state_gather<<<cB * cS * cK, 32, 0, stream>>>(state, ws_qsel, out_state);
}